// NeighborInfluenceModule_6305011991197
// MI455X (gfx1250) — compile-verified
//
#include <hip/hip_runtime.h>
#include <hip/hip_bf16.h>

#define N_NODES 200000
#define D_DIM   128
#define P_PAIRS 4096
#define R_REL   4
#define T_TYPES 3
#define K_NEI   16

typedef __attribute__((ext_vector_type(2))) float v2f;
typedef __attribute__((ext_vector_type(8))) float v8f;

// workspace float offsets
#define WS_WN   0u
#define WS_WS   (WS_WN + T_TYPES * D_DIM)                 // 384
#define WS_SN   (WS_WS + T_TYPES * D_DIM)                 // 768
#define WS_SS   (WS_SN + N_NODES)                          // 200768
#define WS_G    (WS_SS + N_NODES)                          // 400768, size P*R*D

__device__ __forceinline__ float wave_sum32(float v) {
    #pragma unroll
    for (int off = 16; off >= 1; off >>= 1) v += __shfl_xor(v, off, 32);
    return v;
}

// Kernel A: w_n[t*D+d] = sum_e W_phi[t,d,e]*zn[e]; w_s likewise. One wave per (t,d) row.
__global__ void k_fold_zeta(const float* __restrict__ W_phi,
                            const float* __restrict__ W_zeta,
                            float* __restrict__ ws) {
    const int gtid  = blockIdx.x * blockDim.x + threadIdx.x;
    const int row   = gtid >> 5;            // t*D + d
    const int lane  = gtid & 31;
    if (row >= T_TYPES * D_DIM) return;
    const float* __restrict__ wrow = W_phi + (size_t)row * D_DIM;
    float sn = 0.f, ss = 0.f;
    #pragma unroll
    for (int j = 0; j < 4; ++j) {
        const int e = lane + 32 * j;
        const float w = wrow[e];
        sn += w * W_zeta[e];
        ss += w * W_zeta[D_DIM + e];
    }
    sn = wave_sum32(sn);
    ss = wave_sum32(ss);
    if (lane == 0) {
        ws[WS_WN + row] = sn;
        ws[WS_WS + row] = ss;
    }
}

// Kernel B: per-node scores. One wave per node.
__global__ void k_node_scores(const float* __restrict__ embeds,
                              const int* __restrict__ type_ids,
                              float* __restrict__ ws) {
    const int wid_in_blk = threadIdx.x >> 5;
    const int node = blockIdx.x * 8 + wid_in_blk;
    if (node >= N_NODES) return;
    const int lane = threadIdx.x & 31;
    const int t = type_ids[node];
    const float* __restrict__ x  = embeds + (size_t)node * D_DIM;
    const float* __restrict__ wn = ws + WS_WN + t * D_DIM;
    const float* __restrict__ wsv = ws + WS_WS + t * D_DIM;
    float sn = 0.f, ss = 0.f;
    #pragma unroll
    for (int j = 0; j < 4; ++j) {
        const int d = lane + 32 * j;
        const float xv = x[d];
        sn += xv * wn[d];
        ss += xv * wsv[d];
    }
    sn = wave_sum32(sn);
    ss = wave_sum32(ss);
    if (lane == 0) {
        ws[WS_SN + node] = sn;
        ws[WS_SS + node] = ss;
    }
}

// Kernel C: attention softmax + neighbor aggregation, summed over s.
// grid = P blocks of 256 threads; wave w handles (s = w>>2, r = w&3).
// g[p,r,d] = sum_s sum_k alpha[p,s,r,k] * embeds[idx[p,s,r,k], d]
__global__ void k_attend_agg(const float* __restrict__ embeds,
                             const int* __restrict__ node_pairs,
                             const int* __restrict__ neighbor_idx,
                             const float* __restrict__ delta_t,
                             float* __restrict__ ws) {
    __shared__ float lds[8][D_DIM];
    const int p    = blockIdx.x;
    const int wid  = threadIdx.x >> 5;
    const int lane = threadIdx.x & 31;
    const int s = wid >> 2;
    const int r = wid & 3;
    const int k = lane & 15;

    const size_t flat = (((size_t)p * 2 + s) * R_REL + r) * K_NEI + k;
    const int   idx = neighbor_idx[flat];
    const float dt  = delta_t[flat];
    const float en  = ws[WS_SN + idx];
    const int   pn  = node_pairs[p * 2 + s];
    const float es  = ws[WS_SS + pn];
    float e = (en + es) * __expf(-dt);

    // warm the 16 gathered rows while softmax math runs (global_prefetch_b8)
    if (lane < 16) __builtin_prefetch(embeds + (size_t)idx * D_DIM, 0, 0);

    // softmax over K=16 (xor masks 1..8 stay inside each 16-lane half)
    float m = e;
    #pragma unroll
    for (int off = 8; off >= 1; off >>= 1) m = fmaxf(m, __shfl_xor(m, off, 32));
    float a = __expf(e - m);
    float asum = a;
    #pragma unroll
    for (int off = 8; off >= 1; off >>= 1) asum += __shfl_xor(asum, off, 32);
    a = a / asum;

    // aggregate: lane covers 4 consecutive d's -> 512B coalesced row gathers
    const int d0 = lane * 4;
    float acc0 = 0.f, acc1 = 0.f, acc2 = 0.f, acc3 = 0.f;
    #pragma unroll
    for (int kk = 0; kk < K_NEI; ++kk) {
        const float ak = __shfl(a, kk, 32);
        const int   ik = __shfl(idx, kk, 32);
        const float4 v = *(const float4*)(embeds + (size_t)ik * D_DIM + d0);
        acc0 += ak * v.x; acc1 += ak * v.y; acc2 += ak * v.z; acc3 += ak * v.w;
    }
    lds[wid][d0 + 0] = acc0;
    lds[wid][d0 + 1] = acc1;
    lds[wid][d0 + 2] = acc2;
    lds[wid][d0 + 3] = acc3;
    __syncthreads();
    if (s == 0) {
        float* __restrict__ g = ws + WS_G + ((size_t)p * R_REL + r) * D_DIM;
        const float4 o = make_float4(acc0 + lds[wid + 4][d0 + 0],
                                     acc1 + lds[wid + 4][d0 + 1],
                                     acc2 + lds[wid + 4][d0 + 2],
                                     acc3 + lds[wid + 4][d0 + 3]);
        *(float4*)(g + d0) = o;
    }
}

// Kernel D: x_tilde[p,e] = (1/8) sum_r sum_d g[p,r,d] * W_beta_w[r,d,e] + mean_r b[r,e];
// out = sigmoid(x_tilde).  One wave per 16x16 tile, V_WMMA_F32_16X16X4_F32.
__global__ void k_beta_wmma(const float* __restrict__ W_beta_w,
                            const float* __restrict__ W_beta_b,
                            const float* __restrict__ ws,
                            float* __restrict__ out) {
    const int wid  = threadIdx.x >> 5;
    const int lane = threadIdx.x & 31;
    const int tile = blockIdx.x * 8 + wid;     // 2048 tiles total
    const int pt = tile >> 3;                  // 256 M-tiles
    const int et = tile & 7;                   // 8 N-tiles
    const int p0 = pt * 16;
    const int e0 = et * 16;

    const int mn = lane & 15;                  // M (for A) / N (for B) within tile
    const int hi = lane >> 4;                  // 0: K={0,1}, 1: K={2,3} per ISA layout

    v8f c = {0.f, 0.f, 0.f, 0.f, 0.f, 0.f, 0.f, 0.f};

    #pragma unroll
    for (int r = 0; r < R_REL; ++r) {
        const float* __restrict__ ga = ws + WS_G + ((size_t)(p0 + mn) * R_REL + r) * D_DIM;
        const float* __restrict__ bw = W_beta_w + (size_t)r * D_DIM * D_DIM + e0 + mn;
        for (int kk = 0; kk < D_DIM; kk += 4) {
            const int kb = kk + 2 * hi;
            v2f a, b;
            a.x = ga[kb];                      // A[M=mn, K=kb]
            a.y = ga[kb + 1];                  // A[M=mn, K=kb+1]
            b.x = bw[(size_t)kb * D_DIM];      // B[K=kb,   N=mn]
            b.y = bw[(size_t)(kb + 1) * D_DIM];// B[K=kb+1, N=mn]
            c = __builtin_amdgcn_wmma_f32_16x16x4_f32(
                    /*neg_a=*/false, a, /*neg_b=*/false, b,
                    /*c_mod=*/(short)0, c, /*reuse_a=*/false, /*reuse_b=*/false);
        }
    }

    // bias = mean_r W_beta_b[r, e]
    const int e = e0 + mn;
    const float bm = 0.25f * (W_beta_b[e] + W_beta_b[D_DIM + e] +
                              W_beta_b[2 * D_DIM + e] + W_beta_b[3 * D_DIM + e]);

    // D layout: VGPR j holds D[M = j + 8*hi, N = mn]
    #pragma unroll
    for (int j = 0; j < 8; ++j) {
        const int prow = p0 + j + 8 * hi;
        const float x = c[j] * 0.125f + bm;
        out[(size_t)prow * D_DIM + e] = 1.0f / (1.0f + __expf(-x));
    }
}

extern "C" void kernel_launch(void* const* d_in, const int* in_sizes, int n_in,
                              void* d_out, int out_size, void* d_ws, size_t ws_size,
                              hipStream_t stream) {
    const float* node_embeds  = (const float*)d_in[0];
    const int*   node_pairs   = (const int*)d_in[1];
    const int*   node_type_id = (const int*)d_in[2];
    const int*   neighbor_idx = (const int*)d_in[3];
    const float* delta_t      = (const float*)d_in[4];
    const float* W_phi        = (const float*)d_in[5];
    const float* W_zeta       = (const float*)d_in[6];
    const float* W_beta_w     = (const float*)d_in[7];
    const float* W_beta_b     = (const float*)d_in[8];
    float* out = (float*)d_out;
    float* ws  = (float*)d_ws;

    // A: fold zeta into W_phi -> per-type score vectors (384 waves)
    k_fold_zeta<<<48, 256, 0, stream>>>(W_phi, W_zeta, ws);
    // B: per-node attention scores (one wave per node)
    k_node_scores<<<(N_NODES + 7) / 8, 256, 0, stream>>>(node_embeds, node_type_id, ws);
    // C: softmax + gather-aggregate, summed over s (one block per pair)
    k_attend_agg<<<P_PAIRS, 256, 0, stream>>>(node_embeds, node_pairs,
                                              neighbor_idx, delta_t, ws);
    // D: fp32 WMMA GEMM over r + bias + sigmoid (2048 tiles, 8 waves/block)
    k_beta_wmma<<<256, 256, 0, stream>>>(W_beta_w, W_beta_b, ws, out);
}